// NCTSystemSimulator_32332513804674
// MI455X (gfx1250) — compile-verified
//
#include <hip/hip_runtime.h>
#include <math.h>

// Packed 2xfp32 vector -> lowers to VOP3P v_pk_fma_f32 / v_pk_mul_f32 on gfx1250.
typedef float v2f __attribute__((ext_vector_type(2)));

__device__ __forceinline__ v2f vfma(v2f a, v2f b, v2f c) {
  return __builtin_elementwise_fma(a, b, c);
}

// RK4 on ds/dt = A s + c with A = [[0,1],[1,0]] collapses EXACTLY (A^2 = I) to
//   s+ = (ALPHA*I + BETA*A) s + (GAMMA*I + DELTA*A) c,  c = [0, a]
// => x1' = ALPHA x1 + BETA x2 + DELTA*a ; x2' = BETA x1 + ALPHA x2 + GAMMA*a
static constexpr double H_ = 0.01;                                   // DT
static constexpr float ALPHA = (float)(1.0 + H_*H_/2.0 + H_*H_*H_*H_/24.0);
static constexpr float BETA  = (float)(H_ + H_*H_*H_/6.0);
static constexpr float GAMMA = (float)(H_ + H_*H_*H_/6.0);
static constexpr float DELTA = (float)(H_*H_/2.0 + H_*H_*H_*H_/24.0);
// Reward constants: r = -(P1 x1'^2 + P2 x2'^2 + PA u^2) * DT
static constexpr float CP1 = (float)(-1.0  * 0.01);   // -P1*DT
static constexpr float CP2 = (float)(-0.1  * 0.01);   // -P2*DT
static constexpr float CPA = (float)(-0.01 * 0.01);   // -PA*DT
static constexpr float TERM_SCALE = 10.0f;

__device__ __forceinline__ void precompute_sys(
    float b, float w0, float w1, float bias,
    float& m11, float& m12, float& m21, float& m22, float& k1, float& k2) {
  // a = b*u + sin(b), u = w0 x1 + w1 x2 + bias  (u frozen at step start)
  // Fold the linear part of a into the step matrix:
  //   x1' = (ALPHA + DELTA b w0) x1 + (BETA + DELTA b w1) x2 + DELTA*(b*bias + sin b)
  //   x2' = (BETA  + GAMMA b w0) x1 + (ALPHA + GAMMA b w1) x2 + GAMMA*(b*bias + sin b)
  float sb = sinf(b);
  float cb = fmaf(b, bias, sb);
  m11 = fmaf(DELTA * b, w0, ALPHA);
  m12 = fmaf(DELTA * b, w1, BETA);
  m21 = fmaf(GAMMA * b, w0, BETA);
  m22 = fmaf(GAMMA * b, w1, ALPHA);
  k1  = DELTA * cb;
  k2  = GAMMA * cb;
}

__global__ __launch_bounds__(256) void nct_rollout_kernel(
    const float* __restrict__ init_states,  // (B,2) row-major
    const float* __restrict__ b_param,      // (B,1)
    const float* __restrict__ Wp,           // (2,1)
    const float* __restrict__ bias_p,       // (1,)
    const int*   __restrict__ nsteps_p,     // (1,)
    float*       __restrict__ out,          // (B,1)
    int B, int half) {
  const int t = blockIdx.x * blockDim.x + threadIdx.x;
  if (t >= half) return;

  const int  iA   = t;
  const int  iB   = t + half;
  const bool hasB = (iB < B);

  // gfx1250 prefetch path (global_prefetch_b8) — warm the lines we touch.
  __builtin_prefetch(&init_states[2 * iA], 0, 3);
  __builtin_prefetch(&b_param[iA], 0, 3);

  const float w0   = Wp[0];
  const float w1   = Wp[1];
  const float bias = bias_p[0];
  const int   n    = nsteps_p[0];

  // Two independent systems per thread (SoA pack) -> two dep chains, pk-math.
  const float2 sA = ((const float2*)init_states)[iA];
  const float  bA = b_param[iA];
  float2 sB = sA;
  float  bB = bA;
  if (hasB) {
    sB = ((const float2*)init_states)[iB];
    bB = b_param[iB];
  }

  float m11A, m12A, m21A, m22A, k1A, k2A;
  float m11B, m12B, m21B, m22B, k1B, k2B;
  precompute_sys(bA, w0, w1, bias, m11A, m12A, m21A, m22A, k1A, k2A);
  precompute_sys(bB, w0, w1, bias, m11B, m12B, m21B, m22B, k1B, k2B);

  v2f X1  = {sA.x, sB.x};
  v2f X2  = {sA.y, sB.y};
  v2f M11 = {m11A, m11B}, M12 = {m12A, m12B};
  v2f M21 = {m21A, m21B}, M22 = {m22A, m22B};
  v2f K1  = {k1A, k1B},   K2  = {k2A, k2B};
  v2f W0v = {w0, w0}, W1v = {w1, w1}, BIv = {bias, bias};
  v2f C1v = {CP1, CP1}, C2v = {CP2, CP2}, CAv = {CPA, CPA};
  v2f TOT = {0.0f, 0.0f};

#pragma unroll 4
  for (int k = 0; k < n; ++k) {
    // u from pre-step state (matches reference)
    v2f U   = vfma(W0v, X1, vfma(W1v, X2, BIv));
    // exact RK4 step (collapsed affine form)
    v2f NX1 = vfma(M11, X1, vfma(M12, X2, K1));
    v2f NX2 = vfma(M21, X1, vfma(M22, X2, K2));
    // r = -(P1 nx1^2 + P2 nx2^2 + PA u^2)*DT
    v2f R   = vfma(C1v, NX1 * NX1, vfma(C2v, NX2 * NX2, CAv * (U * U)));
    TOT += R;
    X1 = NX1;
    X2 = NX2;
  }

  // terminal: total -= (P1 x1^2 + P2 x2^2) * TERM_SCALE
  const float totA = TOT.x - (1.0f * X1.x * X1.x + 0.1f * X2.x * X2.x) * TERM_SCALE;
  out[iA] = totA;
  if (hasB) {
    const float totB = TOT.y - (1.0f * X1.y * X1.y + 0.1f * X2.y * X2.y) * TERM_SCALE;
    out[iB] = totB;
  }
}

extern "C" void kernel_launch(void* const* d_in, const int* in_sizes, int n_in,
                              void* d_out, int out_size, void* d_ws, size_t ws_size,
                              hipStream_t stream) {
  const float* init_states = (const float*)d_in[0];  // (B,2)
  const float* b_param     = (const float*)d_in[1];  // (B,1)
  const float* W           = (const float*)d_in[2];  // (2,1)
  const float* bias        = (const float*)d_in[3];  // (1,)
  const int*   num_steps   = (const int*)d_in[4];    // scalar
  float*       out         = (float*)d_out;          // (B,1)

  const int B    = in_sizes[0] / 2;
  const int half = (B + 1) / 2;   // two systems per thread
  const int tpb  = 256;           // 8 wave32s per block
  const int nblk = (half + tpb - 1) / tpb;

  nct_rollout_kernel<<<nblk, tpb, 0, stream>>>(init_states, b_param, W, bias,
                                               num_steps, out, B, half);
}